// MolecularGNN_69526930588469
// MI455X (gfx1250) — compile-verified
//
#include <hip/hip_runtime.h>
#include <hip/hip_bf16.h>

#define NN 30000          // nodes
#define EE 480000         // directed edges (before self loops)
#define ET 510000         // EE + NN self loops
#define GG 512            // graphs
#define HIDC 64           // channels per head

typedef float v2f __attribute__((ext_vector_type(2)));
typedef float v8f __attribute__((ext_vector_type(8)));

// ---------- monotonic uint key for float atomic max (handles negatives) ----------
__device__ __forceinline__ unsigned f2key(float f) {
  unsigned u = __float_as_uint(f);
  return (u & 0x80000000u) ? ~u : (u | 0x80000000u);
}
__device__ __forceinline__ float key2f(unsigned k) {
  unsigned u = (k & 0x80000000u) ? (k & 0x7fffffffu) : ~k;
  return __uint_as_float(u);
}
__device__ __forceinline__ bool finite_f(float v) {
  return (__float_as_uint(v) & 0x7f800000u) != 0x7f800000u;
}

// ---------- init kernels ----------
__global__ void fill_f32(float* p, float v, long long n) {
  long long t = (long long)blockIdx.x * blockDim.x + threadIdx.x;
  if (t < n) p[t] = v;
}
__global__ void fill_u32(unsigned* p, unsigned v, long long n) {
  long long t = (long long)blockIdx.x * blockDim.x + threadIdx.x;
  if (t < n) p[t] = v;
}

// ---------- fp32 WMMA GEMM: C[M,Nout] = A[M,K] @ B[K,Nout] (+bias)(+relu) ----------
// One wave per 16x16 C tile; K stepped by 4 via V_WMMA_F32_16X16X4_F32.
// Requires Nout % 16 == 0 and K % 4 == 0 (true for every call site here).
__global__ void wmma_gemm_f32(const float* __restrict__ A, const float* __restrict__ B,
                              const float* __restrict__ bias, float* __restrict__ C,
                              int M, int K, int Nout, int relu) {
  int wave = (int)((blockIdx.x * (long long)blockDim.x + threadIdx.x) >> 5);
  int lane = (int)(threadIdx.x & 31);
  int tilesN = Nout >> 4;
  int tilesM = (M + 15) >> 4;
  if (wave >= tilesM * tilesN) return;          // wave-uniform: EXEC stays all-ones
  int tm = wave / tilesN, tn = wave - tm * tilesN;
  int m0 = tm << 4, n0 = tn << 4;
  int half = lane >> 4;                         // 0: lanes 0-15, 1: lanes 16-31
  int l16  = lane & 15;
  int arow = m0 + l16; if (arow >= M) arow = M - 1;   // clamp (load only)
  int bcol = n0 + l16;

  v8f acc = {};
  for (int k = 0; k < K; k += 4) {
    int ka = k + (half << 1);                   // lanes 0-15: K=k,k+1 ; 16-31: k+2,k+3
    v2f a, b;
    a.x = A[(size_t)arow * K + ka];
    a.y = A[(size_t)arow * K + ka + 1];
    b.x = B[(size_t)ka * Nout + bcol];
    b.y = B[(size_t)(ka + 1) * Nout + bcol];
    acc = __builtin_amdgcn_wmma_f32_16x16x4_f32(false, a, false, b,
                                                (short)0, acc, false, false);
  }
  float bv = bias ? bias[bcol] : 0.0f;
#pragma unroll
  for (int r = 0; r < 8; ++r) {
    int row = m0 + r + (half << 3);             // VGPR r -> M=r (lo lanes) / M=8+r (hi)
    if (row < M) {
      float v = acc[r] + bv;
      if (relu) v = v > 0.0f ? v : 0.0f;
      C[(size_t)row * Nout + bcol] = v;
    }
  }
}

// ---------- per-node attention scores: ssrc/sdst[n,h] = <h[n,h,:], a_src/dst[h,:]> ----------
__global__ void node_scores(const float* __restrict__ h, const float* __restrict__ as_,
                            const float* __restrict__ ad_, float* __restrict__ ssrc,
                            float* __restrict__ sdst, int H) {
  int t = blockIdx.x * blockDim.x + threadIdx.x;
  if (t >= NN * H) return;
  int n = t / H, hh = t - n * H;
  const float* hp  = h   + ((size_t)n * H + hh) * HIDC;
  const float* asp = as_ + hh * HIDC;
  const float* adp = ad_ + hh * HIDC;
  float s1 = 0.0f, s2 = 0.0f;
  for (int c = 0; c < HIDC; ++c) { float v = hp[c]; s1 += v * asp[c]; s2 += v * adp[c]; }
  ssrc[t] = s1; sdst[t] = s2;
}

// ---------- edge pass 1: segment max of leaky_relu over incoming edges ----------
__global__ void edge_max(const int* __restrict__ ei, const float* __restrict__ ssrc,
                         const float* __restrict__ sdst, unsigned* __restrict__ mkey, int H) {
  int t = blockIdx.x * blockDim.x + threadIdx.x;
  if (t >= ET * H) return;
  int e = t / H, hh = t - e * H;
  int s = (e < EE) ? ei[e]      : (e - EE);
  int d = (e < EE) ? ei[EE + e] : (e - EE);
  float v = ssrc[s * H + hh] + sdst[d * H + hh];
  v = v > 0.0f ? v : 0.2f * v;                  // leaky_relu(0.2)
  atomicMax(&mkey[d * H + hh], f2key(v));
}

// ---------- edge pass 2: p = exp(e - m); s[dst,h] += p; acc[dst,h,c] += p * h[src,h,c] ----------
__global__ void edge_accum(const int* __restrict__ ei, const float* __restrict__ ssrc,
                           const float* __restrict__ sdst, const unsigned* __restrict__ mkey,
                           float* __restrict__ ssum, const float* __restrict__ h,
                           float* __restrict__ acc, int H) {
  long long t = (long long)blockIdx.x * blockDim.x + threadIdx.x;
  long long total = (long long)ET * H * HIDC;
  if (t >= total) return;
  int c = (int)(t & (HIDC - 1));
  int eh = (int)(t >> 6);
  int e = eh / H, hh = eh - e * H;
  int s = (e < EE) ? ei[e]      : (e - EE);
  int d = (e < EE) ? ei[EE + e] : (e - EE);
  float v = ssrc[s * H + hh] + sdst[d * H + hh];
  v = v > 0.0f ? v : 0.2f * v;
  float m = key2f(mkey[d * H + hh]);
  float p = expf(v - m);
  if (c == 0) atomicAdd(&ssum[d * H + hh], p);
  atomicAdd(&acc[((size_t)d * H + hh) * HIDC + c],
            p * h[((size_t)s * H + hh) * HIDC + c]);
}

// ---------- node epilogue: divide by softmax sum, +bias, ELU, eval-BN ----------
__global__ void node_finalize(const float* __restrict__ acc, const float* __restrict__ ssum,
                              const float* __restrict__ bias, const float* __restrict__ gamma,
                              const float* __restrict__ beta, float* __restrict__ out, int H) {
  int t = blockIdx.x * blockDim.x + threadIdx.x;
  int F = H * HIDC;
  if (t >= NN * F) return;
  int n = t / F, hc = t - n * F;
  int hh = hc >> 6;
  float s = ssum[n * H + hh];
  float v = acc[t] / (s + 1e-16f) + bias[hc];
  v = v > 0.0f ? v : (expf(v) - 1.0f);                       // ELU
  float gs = gamma[hc] * rsqrtf(1.0f + 1e-5f);               // BN eval (mean 0, var 1)
  out[t] = v * gs + beta[hc];
}

// ---------- graph pooling: mean + max over nodes of each graph ----------
__global__ void pool_accum(const float* __restrict__ h, const int* __restrict__ batch,
                           float* __restrict__ cnt, float* __restrict__ meanacc,
                           unsigned* __restrict__ maxkey) {
  int t = blockIdx.x * blockDim.x + threadIdx.x;
  if (t >= NN * HIDC) return;
  int c = t & (HIDC - 1), n = t >> 6;
  int g = batch[n];
  float v = h[t];
  atomicAdd(&meanacc[g * HIDC + c], v);
  atomicMax(&maxkey[g * HIDC + c], f2key(v));
  if (c == 0) atomicAdd(&cnt[g], 1.0f);
}

__global__ void pool_assemble(const float* __restrict__ cnt, const float* __restrict__ meanacc,
                              const unsigned* __restrict__ maxkey, float* __restrict__ z) {
  int t = blockIdx.x * blockDim.x + threadIdx.x;
  if (t >= GG * 2 * HIDC) return;
  int g = t / (2 * HIDC), c = t - g * 2 * HIDC;
  if (c < HIDC) {
    float cc = cnt[g]; cc = cc > 1.0f ? cc : 1.0f;
    z[t] = meanacc[g * HIDC + c] / cc;
  } else {
    float v = key2f(maxkey[g * HIDC + (c - HIDC)]);
    z[t] = finite_f(v) ? v : 0.0f;                           // empty-graph fallback
  }
}

// ---------- classifier tail: z2[512,32] @ Wc3[32,2] + bc3, log_softmax ----------
__global__ void mlp_out(const float* __restrict__ z2, const float* __restrict__ Wc3,
                        const float* __restrict__ bc3, float* __restrict__ out) {
  int g = blockIdx.x * blockDim.x + threadIdx.x;
  if (g >= GG) return;
  float o0 = bc3[0], o1 = bc3[1];
  for (int k = 0; k < 32; ++k) {
    float v = z2[g * 32 + k];
    o0 += v * Wc3[k * 2];
    o1 += v * Wc3[k * 2 + 1];
  }
  float mx = fmaxf(o0, o1);
  float lse = mx + logf(expf(o0 - mx) + expf(o1 - mx));
  out[g * 2 + 0] = o0 - lse;
  out[g * 2 + 1] = o1 - lse;
}

// ---------- host-side layer driver ----------
static void run_gat_layer(const float* feat, int Kin, int H,
                          const float* W, const float* as_, const float* ad_,
                          const float* bias, const float* gamma, const float* beta,
                          const int* ei, float* buf_h, float* buf_acc,
                          float* ssrc, float* sdst, unsigned* mkey, float* ssum,
                          float* out_feat, hipStream_t stream) {
  const int BS = 256;
  int Nout = H * HIDC;
  long long tiles = (long long)(NN / 16) * (Nout / 16);
  long long gthreads = tiles * 32;
  wmma_gemm_f32<<<(int)((gthreads + BS - 1) / BS), BS, 0, stream>>>(
      feat, W, nullptr, buf_h, NN, Kin, Nout, 0);
  node_scores<<<(NN * H + BS - 1) / BS, BS, 0, stream>>>(buf_h, as_, ad_, ssrc, sdst, H);
  fill_u32<<<(NN * H + BS - 1) / BS, BS, 0, stream>>>(mkey, 0u, (long long)NN * H);
  fill_f32<<<(NN * H + BS - 1) / BS, BS, 0, stream>>>(ssum, 0.0f, (long long)NN * H);
  long long accN = (long long)NN * Nout;
  fill_f32<<<(int)((accN + BS - 1) / BS), BS, 0, stream>>>(buf_acc, 0.0f, accN);
  edge_max<<<(ET * H + BS - 1) / BS, BS, 0, stream>>>(ei, ssrc, sdst, mkey, H);
  long long eaN = (long long)ET * H * HIDC;
  edge_accum<<<(int)((eaN + BS - 1) / BS), BS, 0, stream>>>(ei, ssrc, sdst, mkey, ssum,
                                                            buf_h, buf_acc, H);
  long long nfN = (long long)NN * Nout;
  node_finalize<<<(int)((nfN + BS - 1) / BS), BS, 0, stream>>>(buf_acc, ssum, bias,
                                                               gamma, beta, out_feat, H);
}

extern "C" void kernel_launch(void* const* d_in, const int* in_sizes, int n_in,
                              void* d_out, int out_size, void* d_ws, size_t ws_size,
                              hipStream_t stream) {
  const float* x     = (const float*)d_in[0];
  const int*   ei    = (const int*)d_in[1];
  const int*   batch = (const int*)d_in[2];
  const float* W1  = (const float*)d_in[3];
  const float* a1s = (const float*)d_in[4];
  const float* a1d = (const float*)d_in[5];
  const float* b1  = (const float*)d_in[6];
  const float* g1  = (const float*)d_in[7];
  const float* be1 = (const float*)d_in[8];
  const float* W2  = (const float*)d_in[9];
  const float* a2s = (const float*)d_in[10];
  const float* a2d = (const float*)d_in[11];
  const float* b2  = (const float*)d_in[12];
  const float* g2  = (const float*)d_in[13];
  const float* be2 = (const float*)d_in[14];
  const float* W3  = (const float*)d_in[15];
  const float* a3s = (const float*)d_in[16];
  const float* a3d = (const float*)d_in[17];
  const float* b3  = (const float*)d_in[18];
  const float* g3  = (const float*)d_in[19];
  const float* be3 = (const float*)d_in[20];
  const float* Wc1 = (const float*)d_in[21];
  const float* bc1 = (const float*)d_in[22];
  const float* Wc2 = (const float*)d_in[23];
  const float* bc2 = (const float*)d_in[24];
  const float* Wc3 = (const float*)d_in[25];
  const float* bc3 = (const float*)d_in[26];
  float* out = (float*)d_out;

  // workspace carve-up (~189 MB)
  char* ws = (char*)d_ws;
  size_t off = 0;
  auto carve = [&](size_t bytes) -> void* {
    void* p = ws + off;
    off += (bytes + 255) & ~(size_t)255;
    return p;
  };
  float*    buf_h   = (float*)   carve((size_t)NN * 512 * 4);  // gemm output (h)
  float*    buf_acc = (float*)   carve((size_t)NN * 512 * 4);  // softmax-weighted accumulator
  float*    buf_ft  = (float*)   carve((size_t)NN * 512 * 4);  // finalized features
  float*    ssrc    = (float*)   carve((size_t)NN * 8 * 4);
  float*    sdst    = (float*)   carve((size_t)NN * 8 * 4);
  unsigned* mkey    = (unsigned*)carve((size_t)NN * 8 * 4);
  float*    ssum    = (float*)   carve((size_t)NN * 8 * 4);
  float*    pcnt    = (float*)   carve((size_t)GG * 4);
  float*    pmean   = (float*)   carve((size_t)GG * HIDC * 4);
  unsigned* pmaxk   = (unsigned*)carve((size_t)GG * HIDC * 4);
  float*    zbuf    = (float*)   carve((size_t)GG * 2 * HIDC * 4);
  float*    z1      = (float*)   carve((size_t)GG * HIDC * 4);
  float*    z2      = (float*)   carve((size_t)GG * 32 * 4);
  (void)in_sizes; (void)n_in; (void)out_size; (void)ws_size;

  const int BS = 256;

  // three GAT layers
  run_gat_layer(x,      128, 8, W1, a1s, a1d, b1, g1, be1, ei,
                buf_h, buf_acc, ssrc, sdst, mkey, ssum, buf_ft, stream);
  run_gat_layer(buf_ft, 512, 4, W2, a2s, a2d, b2, g2, be2, ei,
                buf_h, buf_acc, ssrc, sdst, mkey, ssum, buf_ft, stream);
  run_gat_layer(buf_ft, 256, 1, W3, a3s, a3d, b3, g3, be3, ei,
                buf_h, buf_acc, ssrc, sdst, mkey, ssum, buf_ft, stream);

  // global mean + max pooling
  fill_f32<<<(GG + BS - 1) / BS, BS, 0, stream>>>(pcnt, 0.0f, GG);
  fill_f32<<<(GG * HIDC + BS - 1) / BS, BS, 0, stream>>>(pmean, 0.0f, (long long)GG * HIDC);
  fill_u32<<<(GG * HIDC + BS - 1) / BS, BS, 0, stream>>>(pmaxk, 0u, (long long)GG * HIDC);
  pool_accum<<<(NN * HIDC + BS - 1) / BS, BS, 0, stream>>>(buf_ft, batch, pcnt, pmean, pmaxk);
  pool_assemble<<<(GG * 2 * HIDC + BS - 1) / BS, BS, 0, stream>>>(pcnt, pmean, pmaxk, zbuf);

  // classifier MLP: 128 -> 64 (relu) -> 32 (relu) -> 2 (+log_softmax)
  long long t1 = (long long)(GG / 16) * (HIDC / 16) * 32;
  wmma_gemm_f32<<<(int)((t1 + BS - 1) / BS), BS, 0, stream>>>(zbuf, Wc1, bc1, z1,
                                                              GG, 2 * HIDC, HIDC, 1);
  long long t2 = (long long)(GG / 16) * (32 / 16) * 32;
  wmma_gemm_f32<<<(int)((t2 + BS - 1) / BS), BS, 0, stream>>>(z1, Wc2, bc2, z2,
                                                              GG, HIDC, 32, 1);
  mlp_out<<<(GG + BS - 1) / BS, BS, 0, stream>>>(z2, Wc3, bc3, out);
}